// Regularizer_82042465288493
// MI455X (gfx1250) — compile-verified
//
#include <hip/hip_runtime.h>
#include <math.h>

typedef __attribute__((ext_vector_type(2))) float v2f;
typedef __attribute__((ext_vector_type(4))) float v4f;
typedef __attribute__((ext_vector_type(8))) float v8f;

#define SS 256          // samples per batch
#define DD 128          // feature dim
#define NTILE 9         // ceil(130/16) output tiles per dim
#define XL_STRIDE 148   // 148 % 64 == 20 -> conflict-free WMMA operand fetch
#define AUG_STRIDE 132  // 129x(130 used) augmented system
#define NTHREADS 1024   // 32 waves; 1 WG per WGP anyway (LDS-bound)

__launch_bounds__(NTHREADS, 1)
__global__ void locfit_kernel(const float* __restrict__ x,
                              const float* __restrict__ noise,
                              const float* __restrict__ W,
                              const float* __restrict__ bias,
                              float* __restrict__ partial,
                              float inv_total)
{
    __shared__ alignas(16) float XL[SS * XL_STRIDE];     // [s][col]: 0..127 feat, 128=1, 129=y, rest 0
    __shared__ alignas(16) float AUG[129 * AUG_STRIDE];  // [XtX | Xty]
    __shared__ float Ysh[SS];
    __shared__ alignas(16) float Xs[DD];
    __shared__ alignas(16) float Ws[DD];
    __shared__ alignas(16) float Beta[132];
    __shared__ float Red[NTHREADS];

    const int t    = threadIdx.x;
    const int b    = blockIdx.x;
    const int lane = t & 31;
    const int wave = t >> 5;

    if (t < DD) { Xs[t] = x[b * DD + t]; Ws[t] = W[t]; }
    __syncthreads();

    // ---- build xl in LDS: B128 nontemporal global reads, conflict-free B128 LDS writes ----
    const v4f* nrow4 = (const v4f*)(noise + (size_t)b * SS * DD);
    #pragma unroll
    for (int m = 0; m < (SS * DD / 4) / NTHREADS; ++m) {
        int idx = t + m * NTHREADS;
        int s = idx >> 5;                 // 32 float4 per row
        int q = idx & 31;
        v4f nv = __builtin_nontemporal_load(&nrow4[idx]);
        v4f xv = *(const v4f*)(Xs + 4 * q);
        *(v4f*)(XL + s * XL_STRIDE + 4 * q) = xv + 0.1f * nv;
    }
    if (t < SS) {
        XL[t * XL_STRIDE + 128] = 1.0f;
        #pragma unroll
        for (int c2 = 130; c2 < XL_STRIDE; ++c2) XL[t * XL_STRIDE + c2] = 0.0f;
    }
    __syncthreads();

    // ---- inner model y[s] = sigmoid(xl_feat . W + b): 4 threads per sample ----
    {
        int s4 = t >> 2, q4 = t & 3;
        const v4f* row = (const v4f*)(XL + s4 * XL_STRIDE);
        const v4f* wv  = (const v4f*)Ws;
        v4f a = {};
        #pragma unroll
        for (int m = 0; m < 8; ++m) a += row[q4 * 8 + m] * wv[q4 * 8 + m];
        Red[t] = a.x + a.y + a.z + a.w;
    }
    __syncthreads();
    if (t < SS) {
        float z = bias[0] + Red[4*t] + Red[4*t+1] + Red[4*t+2] + Red[4*t+3];
        float y = 1.0f / (1.0f + expf(-z));
        Ysh[t] = y;
        XL[t * XL_STRIDE + 129] = y;      // y as column 129 -> Xty falls out of the Gram pass
    }
    __syncthreads();

    // ---- Gram matrix [XtX | Xty] via V_WMMA_F32_16X16X4_F32, upper triangle + mirror ----
    const int g = lane >> 4;
    const int c = lane & 15;
    for (int p = wave; p < 45; p += 32) {
        int it = 0, rem = p;
        while (rem >= NTILE - it) { rem -= NTILE - it; ++it; }
        int jt = it + rem;
        const int colA = it * 16 + c;
        const int colB = jt * 16 + c;
        v8f acc = {};
        for (int kb = 0; kb < SS; kb += 4) {
            int s0 = kb + 2 * g;                       // K = vgpr + 2*(lane>=16)
            const float* r0 = &XL[s0 * XL_STRIDE];
            const float* r1 = &XL[(s0 + 1) * XL_STRIDE];
            v2f av, bv;
            av.x = r0[colA]; av.y = r1[colA];
            bv.x = r0[colB]; bv.y = r1[colB];
            acc = __builtin_amdgcn_wmma_f32_16x16x4_f32(
                false, av, false, bv, (short)0, acc, false, false);
        }
        #pragma unroll
        for (int r = 0; r < 8; ++r) {
            int i = it * 16 + r + 8 * g;
            int j = jt * 16 + c;
            float v = acc[r];
            if (i <= 128 && j <= 129) AUG[i * AUG_STRIDE + j] = v;
            if (i <= 128 && j <= 128) AUG[j * AUG_STRIDE + i] = v;   // symmetry
        }
    }
    __syncthreads();

    // ---- Gauss-Jordan elimination (SPD, no pivoting): 2-D parallel, 129 barriers ----
    {
        const int tx = t & 31;        // column lanes (consecutive -> conflict-free)
        const int ty = t >> 5;        // row strips
        for (int k = 0; k < 129; ++k) {
            float inv = 1.0f / AUG[k * AUG_STRIDE + k];
            for (int i = ty; i < 129; i += 32) {
                if (i == k) continue;
                float f = AUG[i * AUG_STRIDE + k] * inv;   // wave-broadcast reads
                for (int j = k + 1 + tx; j <= 129; j += 32)
                    AUG[i * AUG_STRIDE + j] -= f * AUG[k * AUG_STRIDE + j];
            }
            __syncthreads();
        }
        if (t < 129) Beta[t] = AUG[t * AUG_STRIDE + 129] / AUG[t * AUG_STRIDE + t];
        __syncthreads();
    }

    // ---- y_lin, BCE, block reduction ----
    {
        int s4 = t >> 2, q4 = t & 3;
        const v4f* row = (const v4f*)(XL + s4 * XL_STRIDE);
        const v4f* bv4 = (const v4f*)Beta;
        v4f a = {};
        #pragma unroll
        for (int m = 0; m < 8; ++m) a += row[q4 * 8 + m] * bv4[q4 * 8 + m];
        Red[t] = a.x + a.y + a.z + a.w;
    }
    __syncthreads();
    float bce = 0.0f;
    if (t < SS) {
        float z = Beta[128] + Red[4*t] + Red[4*t+1] + Red[4*t+2] + Red[4*t+3];
        float y = Ysh[t];
        float l = log1pf(expf(-fabsf(z)));
        bce = -(y * (fminf(z, 0.0f) - l) + (1.0f - y) * (fminf(-z, 0.0f) - l));
    }
    __syncthreads();
    Red[t] = bce;
    __syncthreads();
    for (int off = NTHREADS / 2; off > 0; off >>= 1) {
        if (t < off) Red[t] += Red[t + off];
        __syncthreads();
    }
    if (t == 0) partial[b] = Red[0] * inv_total;
}

__global__ void reduce_kernel(const float* __restrict__ partial,
                              float* __restrict__ out, int n)
{
    __shared__ float red[256];
    int t = threadIdx.x;
    float a = 0.0f;
    for (int i = t; i < n; i += 256) a += partial[i];
    red[t] = a;
    __syncthreads();
    for (int off = 128; off > 0; off >>= 1) {
        if (t < off) red[t] += red[t + off];
        __syncthreads();
    }
    if (t == 0) out[0] = red[0];
}

extern "C" void kernel_launch(void* const* d_in, const int* in_sizes, int n_in,
                              void* d_out, int out_size, void* d_ws, size_t ws_size,
                              hipStream_t stream)
{
    const float* x     = (const float*)d_in[0];   // [B, 128]
    const float* noise = (const float*)d_in[1];   // [B, 256, 128]
    const float* W     = (const float*)d_in[2];   // [128, 1]
    const float* bias  = (const float*)d_in[3];   // [1]
    float* out = (float*)d_out;

    const int B = in_sizes[0] / DD;               // 2048
    float* partial = (float*)d_ws;                // B floats of scratch
    float inv_total = 1.0f / ((float)B * (float)SS);

    locfit_kernel<<<B, NTHREADS, 0, stream>>>(x, noise, W, bias, partial, inv_total);
    reduce_kernel<<<1, 256, 0, stream>>>(partial, out, B);
}